// MSE_loss_1_18030272709297
// MI455X (gfx1250) — compile-verified
//
#include <hip/hip_runtime.h>
#include <hip/hip_bf16.h>

typedef __attribute__((ext_vector_type(2))) float v2f;
typedef __attribute__((ext_vector_type(8))) float v8f;

#define HW_ELEMS   4194304      // 2048*2048
#define NCH        9
#define NB         64           // blocks per channel
#define NTHREADS   256
#define F4_PER_CH  1048576      // HW_ELEMS / 4
#define F4_STRIDE  16384        // NB * NTHREADS
#define NITER      64           // F4_PER_CH / F4_STRIDE
#define NOUTER     32           // NITER / 2 (manual unroll x2)
#define NEG_BIG    (-3.402823466e38f)

// Sum of ALL 256 slots of the 16x16 accumulator across the wave = 16x the true
// accumulated total (every row of D is identical when A == ones). Layout-proof.
__device__ __forceinline__ float wmma_wave_total(v8f c) {
    float v = c[0] + c[1] + c[2] + c[3] + c[4] + c[5] + c[6] + c[7];
    #pragma unroll
    for (int m = 16; m >= 1; m >>= 1) v += __shfl_xor(v, m, 32);
    return v * 0.0625f;  // /16
}

__device__ __forceinline__ v8f wmma_acc(v8f c, float s0, float s1) {
    v2f ones; ones[0] = 1.0f; ones[1] = 1.0f;
    v2f b;    b[0] = s0;      b[1] = s1;
    return __builtin_amdgcn_wmma_f32_16x16x4_f32(
        /*neg_a=*/false, ones, /*neg_b=*/false, b,
        /*c_mod=*/(short)0, c, /*reuse_a=*/false, /*reuse_b=*/false);
}

__device__ __forceinline__ void top10_insert(float* t, float x) {
    if (x > t[9]) {
        #pragma unroll
        for (int j = 0; j < 10; ++j) {
            if (x > t[j]) { float tmp = t[j]; t[j] = x; x = tmp; }
        }
    }
}

// ---------------- Pass 1: per-channel sum + per-block top-10 -----------------
__global__ void k_stats1(const float* __restrict__ pattern,
                         float* __restrict__ ws_sums,   // [NCH][NB]
                         float* __restrict__ ws_top) {  // [NCH][NB][10]
    const int c   = blockIdx.y;
    const int bx  = blockIdx.x;
    const int tid = threadIdx.x;

    const float4* p4 = (const float4*)(pattern + (size_t)c * HW_ELEMS);

    float t[10];
    #pragma unroll
    for (int j = 0; j < 10; ++j) t[j] = NEG_BIG;

    v8f acc0 = {};
    v8f acc1 = {};
    int idx = bx * NTHREADS + tid;
    for (int it = 0; it < NOUTER; ++it, idx += 2 * F4_STRIDE) {
        // two independent b128 loads in flight before any use
        float4 va = p4[idx];
        float4 vb = p4[idx + F4_STRIDE];
        // independent accumulators: no WMMA->WMMA RAW stall
        acc0 = wmma_acc(acc0, va.x, va.y);
        acc1 = wmma_acc(acc1, va.z, va.w);
        acc0 = wmma_acc(acc0, vb.x, vb.y);
        acc1 = wmma_acc(acc1, vb.z, vb.w);
        // divergent inserts; EXEC reconverges before next iteration's WMMA
        top10_insert(t, va.x);
        top10_insert(t, va.y);
        top10_insert(t, va.z);
        top10_insert(t, va.w);
        top10_insert(t, vb.x);
        top10_insert(t, vb.y);
        top10_insert(t, vb.z);
        top10_insert(t, vb.w);
    }

    __shared__ float cand[NTHREADS * 10];
    __shared__ float sVal[NTHREADS];
    __shared__ int   sIdx[NTHREADS];
    __shared__ float wsum[NTHREADS / 32];

    #pragma unroll
    for (int j = 0; j < 10; ++j) cand[tid * 10 + j] = t[j];

    float wtot = wmma_wave_total(acc0 + acc1);
    if ((tid & 31) == 0) wsum[tid >> 5] = wtot;
    __syncthreads();

    // 10 rounds of parallel max-extract over 2560 candidates
    for (int r = 0; r < 10; ++r) {
        float best = NEG_BIG; int bidx = 0;
        for (int i = tid; i < NTHREADS * 10; i += NTHREADS) {
            float v = cand[i];
            if (v > best) { best = v; bidx = i; }
        }
        sVal[tid] = best; sIdx[tid] = bidx;
        __syncthreads();
        for (int st = NTHREADS / 2; st > 0; st >>= 1) {
            if (tid < st && sVal[tid + st] > sVal[tid]) {
                sVal[tid] = sVal[tid + st]; sIdx[tid] = sIdx[tid + st];
            }
            __syncthreads();
        }
        if (tid == 0) {
            ws_top[(c * NB + bx) * 10 + r] = sVal[0];
            cand[sIdx[0]] = NEG_BIG;
        }
        __syncthreads();
    }

    if (tid == 0) {
        float s = 0.f;
        #pragma unroll
        for (int i = 0; i < NTHREADS / 32; ++i) s += wsum[i];
        ws_sums[c * NB + bx] = s;
    }
}

// ---------------- Pass 1b: merge per-channel stats ---------------------------
__global__ void k_stats2(const float* __restrict__ ws_sums,
                         const float* __restrict__ ws_top,
                         float* __restrict__ ws_stats) {  // [NCH][2] = mean, 1/denom
    const int c   = blockIdx.x;
    const int tid = threadIdx.x;

    __shared__ float cand[NB * 10];
    __shared__ float sVal[NTHREADS];
    __shared__ int   sIdx[NTHREADS];
    __shared__ float sh_total;

    for (int i = tid; i < NB * 10; i += NTHREADS) cand[i] = ws_top[c * NB * 10 + i];

    float s = (tid < NB) ? ws_sums[c * NB + tid] : 0.f;
    sVal[tid] = s;
    __syncthreads();
    for (int st = NTHREADS / 2; st > 0; st >>= 1) {
        if (tid < st) sVal[tid] += sVal[tid + st];
        __syncthreads();
    }
    if (tid == 0) sh_total = sVal[0];
    __syncthreads();

    const float mean = sh_total * (1.0f / (float)HW_ELEMS);

    float topsum = 0.f;  // only tid==0's copy matters
    for (int r = 0; r < 10; ++r) {
        float best = NEG_BIG; int bidx = 0;
        for (int i = tid; i < NB * 10; i += NTHREADS) {
            float v = cand[i];
            if (v > best) { best = v; bidx = i; }
        }
        sVal[tid] = best; sIdx[tid] = bidx;
        __syncthreads();
        for (int st = NTHREADS / 2; st > 0; st >>= 1) {
            if (tid < st && sVal[tid + st] > sVal[tid]) {
                sVal[tid] = sVal[tid + st]; sIdx[tid] = sIdx[tid + st];
            }
            __syncthreads();
        }
        if (tid == 0) { topsum += sVal[0]; cand[sIdx[0]] = NEG_BIG; }
        __syncthreads();
    }

    if (tid == 0) {
        // top10(x - mean) == top10(x) - mean  =>  denom = mean(top10(x)) - mean
        float denom = topsum * 0.1f - mean;
        if (denom < 1e-20f) denom += 1e-19f;
        ws_stats[c * 2 + 0] = mean;
        ws_stats[c * 2 + 1] = 1.0f / denom;
    }
}

// ---------------- Pass 2: masked MSE, WMMA-accumulated -----------------------
__global__ void k_loss(const float* __restrict__ pattern,
                       const float* __restrict__ gt,
                       const float* __restrict__ mask,
                       const float* __restrict__ ws_stats,
                       float* __restrict__ ws_loss) {  // [NCH][NB]
    const int c   = blockIdx.y;
    const int bx  = blockIdx.x;
    const int tid = threadIdx.x;

    const float mean = ws_stats[c * 2 + 0];
    const float inv  = ws_stats[c * 2 + 1];

    const float4* p4 = (const float4*)(pattern + (size_t)c * HW_ELEMS);
    const float4* g4 = (const float4*)gt;
    const float4* m4 = (const float4*)mask;

    v8f acc0 = {};
    v8f acc1 = {};
    int idx = bx * NTHREADS + tid;
    for (int it = 0; it < NOUTER; ++it, idx += 2 * F4_STRIDE) {
        const int idx2 = idx + F4_STRIDE;
        __builtin_prefetch(p4 + idx + 2 * F4_STRIDE, 0, 0);
        __builtin_prefetch(g4 + idx + 2 * F4_STRIDE, 0, 0);
        __builtin_prefetch(m4 + idx + 2 * F4_STRIDE, 0, 0);
        // six b128 loads in flight before any use
        float4 xa = p4[idx];
        float4 ga = g4[idx];
        float4 ma = m4[idx];
        float4 xb = p4[idx2];
        float4 gb = g4[idx2];
        float4 mb = m4[idx2];

        float a0 = ((xa.x - mean) * inv - ga.x) * ma.x;
        float a1 = ((xa.y - mean) * inv - ga.y) * ma.y;
        float a2 = ((xa.z - mean) * inv - ga.z) * ma.z;
        float a3 = ((xa.w - mean) * inv - ga.w) * ma.w;
        float b0 = ((xb.x - mean) * inv - gb.x) * mb.x;
        float b1 = ((xb.y - mean) * inv - gb.y) * mb.y;
        float b2 = ((xb.z - mean) * inv - gb.z) * mb.z;
        float b3 = ((xb.w - mean) * inv - gb.w) * mb.w;

        acc0 = wmma_acc(acc0, a0 * a0, a1 * a1);
        acc1 = wmma_acc(acc1, a2 * a2, a3 * a3);
        acc0 = wmma_acc(acc0, b0 * b0, b1 * b1);
        acc1 = wmma_acc(acc1, b2 * b2, b3 * b3);
    }

    __shared__ float wsum[NTHREADS / 32];
    float wtot = wmma_wave_total(acc0 + acc1);
    if ((tid & 31) == 0) wsum[tid >> 5] = wtot;
    __syncthreads();
    if (tid == 0) {
        float s = 0.f;
        #pragma unroll
        for (int i = 0; i < NTHREADS / 32; ++i) s += wsum[i];
        ws_loss[c * NB + bx] = s;
    }
}

// ---------------- Final deterministic reduction ------------------------------
__global__ void k_final(const float* __restrict__ ws_loss, float* __restrict__ out) {
    __shared__ float sVal[NTHREADS];
    const int tid = threadIdx.x;
    float s = 0.f;
    for (int i = tid; i < NCH * NB; i += NTHREADS) s += ws_loss[i];
    sVal[tid] = s;
    __syncthreads();
    for (int st = NTHREADS / 2; st > 0; st >>= 1) {
        if (tid < st) sVal[tid] += sVal[tid + st];
        __syncthreads();
    }
    if (tid == 0) out[0] = sVal[0] * (1.0f / (float)HW_ELEMS);
}

extern "C" void kernel_launch(void* const* d_in, const int* in_sizes, int n_in,
                              void* d_out, int out_size, void* d_ws, size_t ws_size,
                              hipStream_t stream) {
    const float* pattern = (const float*)d_in[0];  // [1,9,2048,2048]
    const float* gt      = (const float*)d_in[1];  // [2048,2048]
    const float* mask    = (const float*)d_in[2];  // [2048,2048]

    float* ws       = (float*)d_ws;
    float* ws_sums  = ws;                    // 9*64     = 576
    float* ws_top   = ws + 576;              // 9*64*10  = 5760
    float* ws_stats = ws + 576 + 5760;       // 9*2      = 18
    float* ws_loss  = ws + 576 + 5760 + 32;  // 9*64     = 576

    dim3 grid(NB, NCH);
    k_stats1<<<grid, NTHREADS, 0, stream>>>(pattern, ws_sums, ws_top);
    k_stats2<<<NCH, NTHREADS, 0, stream>>>(ws_sums, ws_top, ws_stats);
    k_loss  <<<grid, NTHREADS, 0, stream>>>(pattern, gt, mask, ws_stats, ws_loss);
    k_final <<<1, NTHREADS, 0, stream>>>(ws_loss, (float*)d_out);
}